// MultiGraphDeconv_37855841746992
// MI455X (gfx1250) — compile-verified
//
#include <hip/hip_runtime.h>

#define NN     2048   // graph nodes
#define TOPK   512
#define KORD   4
#define OUTC   256
#define BATCH  2048
#define BS     32     // Jacobi block size
#define NB     64     // NN / BS
#define LOC    64     // 2*BS local subproblem
#define BSWEEPS 8     // block-Jacobi sweeps
#define LSWEEPS 6     // local 64x64 Jacobi sweeps

typedef __attribute__((ext_vector_type(2))) float v2f;
typedef __attribute__((ext_vector_type(8))) float v8f;

// CDNA5 FP32 WMMA: D(16x16) = A(16x4) * B(4x16) + C, full FP32 precision.
__device__ __forceinline__ v8f wmma_f32(v2f a, v2f b, v8f c) {
  return __builtin_amdgcn_wmma_f32_16x16x4_f32(false, a, false, b, (short)0, c,
                                               false, false);
}

// round-robin (circle method) pairing: n even, rounds 0..n-2, pairs 0..n/2-1
__device__ __forceinline__ int rr_pos(int i, int r, int n) {
  return (i == 0) ? 0 : (1 + (i - 1 + r) % (n - 1));
}
// local index (0..63) -> global row/col for block pair (bp,bq)
__device__ __forceinline__ int grow(int l, int bp, int bq) {
  return (l < BS) ? (bp * BS + l) : (bq * BS + (l - BS));
}

// ---------------- Laplacian setup ----------------
__global__ void degree_kernel(const float* __restrict__ g, float* __restrict__ dinv) {
  __shared__ float red[256];
  int i = blockIdx.x;
  float s = 0.f;
  for (int j = threadIdx.x; j < NN; j += 256) s += g[(size_t)i * NN + j];
  red[threadIdx.x] = s;
  __syncthreads();
  for (int off = 128; off > 0; off >>= 1) {
    if (threadIdx.x < off) red[threadIdx.x] += red[threadIdx.x + off];
    __syncthreads();
  }
  if (threadIdx.x == 0) dinv[i] = 1.0f / sqrtf(red[0]);
}

__global__ void init_lu_kernel(const float* __restrict__ g, const float* __restrict__ dinv,
                               float* __restrict__ A, float* __restrict__ U) {
  int i = blockIdx.x;
  float di = dinv[i];
  for (int j = threadIdx.x; j < NN; j += blockDim.x) {
    float l = ((i == j) ? 1.f : 0.f) - di * g[(size_t)i * NN + j] * dinv[j];
    A[(size_t)i * NN + j] = l;
    U[(size_t)i * NN + j] = (i == j) ? 1.f : 0.f;
  }
}

// ---------------- block Jacobi: local 64x64 solve (scalar, in LDS) ----------------
__global__ void jacobi_block_solve(const float* __restrict__ A, float* __restrict__ Vws,
                                   int round) {
  __shared__ float S[LOC][LOC + 1];
  __shared__ float Vv[LOC][LOC + 1];
  __shared__ float cc[LOC / 2], ss[LOC / 2];
  int pairk = blockIdx.x;
  int bp = rr_pos(pairk, round, NB);
  int bq = rr_pos(NB - 1 - pairk, round, NB);
  int tid = threadIdx.x;  // 256
  for (int idx = tid; idx < LOC * LOC; idx += blockDim.x) {
    int i = idx >> 6, j = idx & 63;
    S[i][j] = A[(size_t)grow(i, bp, bq) * NN + grow(j, bp, bq)];
    Vv[i][j] = (i == j) ? 1.f : 0.f;
  }
  __syncthreads();
  for (int sweep = 0; sweep < LSWEEPS; ++sweep) {
    for (int r = 0; r < LOC - 1; ++r) {
      if (tid < LOC / 2) {
        int p = rr_pos(tid, r, LOC), q = rr_pos(LOC - 1 - tid, r, LOC);
        float app = S[p][p], aqq = S[q][q], apq = S[p][q];
        float c = 1.f, s = 0.f;
        if (fabsf(apq) > 1e-30f) {
          float tau = (aqq - app) / (2.f * apq);
          float tt = copysignf(1.f, tau) / (fabsf(tau) + sqrtf(1.f + tau * tau));
          c = 1.f / sqrtf(1.f + tt * tt);
          s = tt * c;
        }
        cc[tid] = c;
        ss[tid] = s;
      }
      __syncthreads();
      // rows: S <- G^T S   (disjoint row pairs -> race-free in place)
      for (int idx = tid; idx < (LOC / 2) * LOC; idx += blockDim.x) {
        int kk = idx >> 6, j = idx & 63;
        int p = rr_pos(kk, r, LOC), q = rr_pos(LOC - 1 - kk, r, LOC);
        float c = cc[kk], s = ss[kk];
        float sp = S[p][j], sq = S[q][j];
        S[p][j] = c * sp - s * sq;
        S[q][j] = s * sp + c * sq;
      }
      __syncthreads();
      // cols: S <- S G ; V <- V G
      for (int idx = tid; idx < (LOC / 2) * LOC; idx += blockDim.x) {
        int kk = idx >> 6, i = idx & 63;
        int p = rr_pos(kk, r, LOC), q = rr_pos(LOC - 1 - kk, r, LOC);
        float c = cc[kk], s = ss[kk];
        float sp = S[i][p], sq = S[i][q];
        S[i][p] = c * sp - s * sq;
        S[i][q] = s * sp + c * sq;
        float vp = Vv[i][p], vq = Vv[i][q];
        Vv[i][p] = c * vp - s * vq;
        Vv[i][q] = s * vp + c * vq;
      }
      __syncthreads();
    }
  }
  for (int idx = tid; idx < LOC * LOC; idx += blockDim.x)
    Vws[(size_t)pairk * LOC * LOC + idx] = Vv[idx >> 6][idx & 63];
}

// ---------------- block Jacobi: A[rows(p,q),:] <- V^T A[rows(p,q),:]  (WMMA) ----------------
__global__ void jacobi_apply_left(float* __restrict__ A, const float* __restrict__ Vws,
                                  int round) {
  __shared__ float Vs[LOC][LOC + 1];
  __shared__ float Ins[LOC][16];
  int pairk = blockIdx.x;      // 0..31
  int stripe = blockIdx.y;     // 0..127 : 16-wide column stripe
  int bp = rr_pos(pairk, round, NB);
  int bq = rr_pos(NB - 1 - pairk, round, NB);
  int tid = threadIdx.x;       // 128 = 4 waves
  for (int idx = tid; idx < LOC * LOC; idx += 128)
    Vs[idx >> 6][idx & 63] = Vws[(size_t)pairk * LOC * LOC + idx];
  for (int idx = tid; idx < LOC * 16; idx += 128) {
    int i = idx >> 4, j = idx & 15;
    Ins[i][j] = A[(size_t)grow(i, bp, bq) * NN + stripe * 16 + j];
  }
  __syncthreads();
  int wave = tid >> 5, lane = tid & 31;
  int lo = lane & 15, hi = lane >> 4;
  v8f acc = {};
  for (int kk = 0; kk < LOC; kk += 4) {
    v2f a, b;
    a.x = Vs[kk + 2 * hi][wave * 16 + lo];      // (V^T)[m,k] = V[k,m]
    a.y = Vs[kk + 2 * hi + 1][wave * 16 + lo];
    b.x = Ins[kk + 2 * hi][lo];
    b.y = Ins[kk + 2 * hi + 1][lo];
    acc = wmma_f32(a, b, acc);
  }
  for (int v = 0; v < 8; ++v) {
    int lrow = wave * 16 + v + 8 * hi;
    A[(size_t)grow(lrow, bp, bq) * NN + stripe * 16 + lo] = acc[v];
  }
}

// ------------- block Jacobi: A[:,cols(p,q)] <- A[:,cols] V ; U likewise  (WMMA) -------------
__global__ void jacobi_apply_right(float* __restrict__ A, float* __restrict__ U,
                                   const float* __restrict__ Vws, int round) {
  __shared__ float Vs[LOC][LOC + 1];
  __shared__ float Sa[16][LOC];
  __shared__ float Su[16][LOC];
  int pairk = blockIdx.x;      // 0..31
  int stripe = blockIdx.y;     // 0..127 : 16-row stripe
  int bp = rr_pos(pairk, round, NB);
  int bq = rr_pos(NB - 1 - pairk, round, NB);
  int tid = threadIdx.x;       // 128
  for (int idx = tid; idx < LOC * LOC; idx += 128)
    Vs[idx >> 6][idx & 63] = Vws[(size_t)pairk * LOC * LOC + idx];
  for (int idx = tid; idx < 16 * LOC; idx += 128) {
    int i = idx >> 6, j = idx & 63;
    int gc = grow(j, bp, bq);
    Sa[i][j] = A[(size_t)(stripe * 16 + i) * NN + gc];
    Su[i][j] = U[(size_t)(stripe * 16 + i) * NN + gc];
  }
  __syncthreads();
  int wave = tid >> 5, lane = tid & 31;
  int lo = lane & 15, hi = lane >> 4;
  v8f accA = {}, accU = {};
  for (int kk = 0; kk < LOC; kk += 4) {
    v2f a, au, b;
    a.x = Sa[lo][kk + 2 * hi];
    a.y = Sa[lo][kk + 2 * hi + 1];
    au.x = Su[lo][kk + 2 * hi];
    au.y = Su[lo][kk + 2 * hi + 1];
    b.x = Vs[kk + 2 * hi][wave * 16 + lo];
    b.y = Vs[kk + 2 * hi + 1][wave * 16 + lo];
    accA = wmma_f32(a, b, accA);
    accU = wmma_f32(au, b, accU);
  }
  for (int v = 0; v < 8; ++v) {
    int row = stripe * 16 + v + 8 * hi;
    int gc = grow(wave * 16 + lo, bp, bq);
    A[(size_t)row * NN + gc] = accA[v];
    U[(size_t)row * NN + gc] = accU[v];
  }
}

// ---------------- eigenvalue extraction / top-k selection ----------------
__global__ void extract_diag_kernel(const float* __restrict__ A, float* __restrict__ lam) {
  int i = blockIdx.x * blockDim.x + threadIdx.x;
  if (i < NN) lam[i] = A[(size_t)i * NN + i];
}

__global__ void rank_select_kernel(const float* __restrict__ lam, int* __restrict__ perm,
                                   float* __restrict__ lam_sorted) {
  __shared__ int red[256];
  int i = blockIdx.x;
  float li = lam[i];
  int cnt = 0;
  for (int j = threadIdx.x; j < NN; j += 256) {
    float lj = lam[j];
    cnt += (lj > li) || (lj == li && j < i);   // descending, stable
  }
  red[threadIdx.x] = cnt;
  __syncthreads();
  for (int off = 128; off > 0; off >>= 1) {
    if (threadIdx.x < off) red[threadIdx.x] += red[threadIdx.x + off];
    __syncthreads();
  }
  if (threadIdx.x == 0 && red[0] < TOPK) {
    perm[red[0]] = i;
    lam_sorted[red[0]] = li;
  }
}

// Chebyshev coefficients T_k(d_t); min/max over full diag matrix (off-diag zeros included)
__global__ void cheb_kernel(const float* __restrict__ lam_sorted, float* __restrict__ cheb) {
  __shared__ float mnr[TOPK], mxr[TOPK];
  int t = threadIdx.x;  // 512
  float v = lam_sorted[t];
  mnr[t] = fminf(v, 0.f);
  mxr[t] = fmaxf(v, 0.f);
  __syncthreads();
  for (int off = TOPK / 2; off > 0; off >>= 1) {
    if (t < off) {
      mnr[t] = fminf(mnr[t], mnr[t + off]);
      mxr[t] = fmaxf(mxr[t], mxr[t + off]);
    }
    __syncthreads();
  }
  float mn = mnr[0], mx = mxr[0];
  float d = (mx == mn) ? 0.f : (2.f * (v - mn) / (mx - mn) - 1.f);
  float t0 = 1.f, t1 = d;
  float t2 = 2.f * d * t1 - t0;
  float t3 = 2.f * d * t2 - t1;
  cheb[t * KORD + 0] = t0;
  cheb[t * KORD + 1] = t1;
  cheb[t * KORD + 2] = t2;
  cheb[t * KORD + 3] = t3;
}

// Up[t][n] = U[n][perm[t]]  (Usel^T, row-major 512x2048; U is L2-resident)
__global__ void gather_up_kernel(const float* __restrict__ U, const int* __restrict__ perm,
                                 float* __restrict__ Up) {
  int t = blockIdx.x;
  int col = perm[t];
  for (int n = threadIdx.x; n < NN; n += blockDim.x)
    Up[(size_t)t * NN + n] = U[(size_t)n * NN + col];
}

// W_eff[t,c] = sum_k T_k(d_t) * sum_n Up[t,n] * W[k,n,c]   (WMMA, row-scaled A)
__global__ void weff_gemm_kernel(const float* __restrict__ Up, const float* __restrict__ W,
                                 const float* __restrict__ cheb, float* __restrict__ Weff) {
  __shared__ float At[16][33];
  __shared__ float Bt[32][65];
  int rt = blockIdx.x;  // 0..31 row tile
  int cg = blockIdx.y;  // 0..3  64-col group
  int tid = threadIdx.x;  // 128
  int wave = tid >> 5, lane = tid & 31;
  int lo = lane & 15, hi = lane >> 4;
  float ch[KORD];
  for (int k = 0; k < KORD; ++k) ch[k] = cheb[(rt * 16 + lo) * KORD + k];
  v8f acc = {};
  for (int kb = 0; kb < NN; kb += 32) {
    for (int idx = tid; idx < 16 * 32; idx += 128) {
      int i = idx >> 5, j = idx & 31;
      At[i][j] = Up[(size_t)(rt * 16 + i) * NN + kb + j];
    }
    for (int k = 0; k < KORD; ++k) {
      __syncthreads();
      for (int idx = tid; idx < 32 * 64; idx += 128) {
        int i = idx >> 6, j = idx & 63;
        Bt[i][j] = W[((size_t)k * NN + kb + i) * OUTC + cg * 64 + j];
      }
      __syncthreads();
      for (int c0 = 0; c0 < 32; c0 += 4) {
        v2f a, b;
        a.x = At[lo][c0 + 2 * hi] * ch[k];
        a.y = At[lo][c0 + 2 * hi + 1] * ch[k];
        b.x = Bt[c0 + 2 * hi][wave * 16 + lo];
        b.y = Bt[c0 + 2 * hi + 1][wave * 16 + lo];
        acc = wmma_f32(a, b, acc);
      }
    }
    __syncthreads();
  }
  for (int v = 0; v < 8; ++v)
    Weff[(size_t)(rt * 16 + v + 8 * hi) * OUTC + cg * 64 + wave * 16 + lo] = acc[v];
}

// raw[b,c] = inputs[b,:] @ W_eff[:,c] + bias[c]   (WMMA)
__global__ void final_gemm_kernel(const float* __restrict__ inp, const float* __restrict__ Weff,
                                  const float* __restrict__ bias, float* __restrict__ raw) {
  __shared__ float At[16][33];
  __shared__ float Bt[32][65];
  int rt = blockIdx.x;  // 0..127
  int cg = blockIdx.y;  // 0..3
  int tid = threadIdx.x;
  int wave = tid >> 5, lane = tid & 31;
  int lo = lane & 15, hi = lane >> 4;
  v8f acc = {};
  for (int kb = 0; kb < TOPK; kb += 32) {
    for (int idx = tid; idx < 16 * 32; idx += 128) {
      int i = idx >> 5, j = idx & 31;
      At[i][j] = inp[(size_t)(rt * 16 + i) * TOPK + kb + j];
    }
    for (int idx = tid; idx < 32 * 64; idx += 128) {
      int i = idx >> 6, j = idx & 63;
      Bt[i][j] = Weff[(size_t)(kb + i) * OUTC + cg * 64 + j];
    }
    __syncthreads();
    for (int c0 = 0; c0 < 32; c0 += 4) {
      v2f a, b;
      a.x = At[lo][c0 + 2 * hi];
      a.y = At[lo][c0 + 2 * hi + 1];
      b.x = Bt[c0 + 2 * hi][wave * 16 + lo];
      b.y = Bt[c0 + 2 * hi + 1][wave * 16 + lo];
      acc = wmma_f32(a, b, acc);
    }
    __syncthreads();
  }
  for (int v = 0; v < 8; ++v) {
    int col = cg * 64 + wave * 16 + lo;
    raw[(size_t)(rt * 16 + v + 8 * hi) * OUTC + col] = acc[v] + bias[col];
  }
}

// per-row min-max rescale to [-1,1]
__global__ void rescale_kernel(const float* __restrict__ raw, float* __restrict__ out) {
  __shared__ float mnr[256], mxr[256];
  int b = blockIdx.x;
  int t = threadIdx.x;  // 256 == OUTC
  float v = raw[(size_t)b * OUTC + t];
  mnr[t] = v;
  mxr[t] = v;
  __syncthreads();
  for (int off = 128; off > 0; off >>= 1) {
    if (t < off) {
      mnr[t] = fminf(mnr[t], mnr[t + off]);
      mxr[t] = fmaxf(mxr[t], mxr[t + off]);
    }
    __syncthreads();
  }
  float mn = mnr[0], mx = mxr[0];
  out[(size_t)b * OUTC + t] = (mx == mn) ? v : (2.f * (v - mn) / (mx - mn) - 1.f);
}

extern "C" void kernel_launch(void* const* d_in, const int* in_sizes, int n_in,
                              void* d_out, int out_size, void* d_ws, size_t ws_size,
                              hipStream_t stream) {
  (void)in_sizes; (void)n_in; (void)out_size; (void)ws_size;
  const float* inp  = (const float*)d_in[0];  // [B,1,TOPK]
  const float* g    = (const float*)d_in[1];  // [N,N]
  const float* w    = (const float*)d_in[2];  // [K,1,N,OUTC]
  const float* bias = (const float*)d_in[3];  // [1,1,OUTC]
  float* out = (float*)d_out;

  float* ws = (float*)d_ws;
  size_t off = 0;
  float* Amat = ws + off; off += (size_t)NN * NN;            // 16 MB
  float* Umat = ws + off; off += (size_t)NN * NN;            // 16 MB
  float* Vws  = ws + off; off += (size_t)(NB / 2) * LOC * LOC;
  float* dinv = ws + off; off += NN;
  float* lam  = ws + off; off += NN;
  float* lams = ws + off; off += TOPK;
  int*   perm = (int*)(ws + off); off += TOPK;
  float* cheb = ws + off; off += (size_t)TOPK * KORD;
  float* Up   = ws + off; off += (size_t)TOPK * NN;          // 4 MB
  float* Weff = ws + off; off += (size_t)TOPK * OUTC;
  float* raw  = ws + off; off += (size_t)BATCH * OUTC;

  // L = I - D^{-1/2} A D^{-1/2}; U = I
  degree_kernel<<<NN, 256, 0, stream>>>(g, dinv);
  init_lu_kernel<<<NN, 256, 0, stream>>>(g, dinv, Amat, Umat);

  // block two-sided Jacobi eigensolver: heavy work = WMMA FP32 rotation GEMMs
  for (int sweep = 0; sweep < BSWEEPS; ++sweep) {
    for (int r = 0; r < NB - 1; ++r) {
      jacobi_block_solve<<<NB / 2, 256, 0, stream>>>(Amat, Vws, r);
      jacobi_apply_left<<<dim3(NB / 2, NN / 16), 128, 0, stream>>>(Amat, Vws, r);
      jacobi_apply_right<<<dim3(NB / 2, NN / 16), 128, 0, stream>>>(Amat, Umat, Vws, r);
    }
  }

  // top-512 eigenpairs (descending), Chebyshev coefficients
  extract_diag_kernel<<<NN / 256, 256, 0, stream>>>(Amat, lam);
  rank_select_kernel<<<NN, 256, 0, stream>>>(lam, perm, lams);
  cheb_kernel<<<1, TOPK, 0, stream>>>(lams, cheb);
  gather_up_kernel<<<TOPK, 256, 0, stream>>>(Umat, perm, Up);

  // fold whole tail into one effective weight, then single GEMM + rescale
  weff_gemm_kernel<<<dim3(TOPK / 16, OUTC / 64), 128, 0, stream>>>(Up, w, cheb, Weff);
  final_gemm_kernel<<<dim3(BATCH / 16, OUTC / 64), 128, 0, stream>>>(inp, Weff, bias, raw);
  rescale_kernel<<<BATCH, 256, 0, stream>>>(raw, out);
}